// GAT_73478300500626
// MI455X (gfx1250) — compile-verified
//
#include <hip/hip_runtime.h>
#include <hip/hip_bf16.h>
#include <cfloat>

typedef __attribute__((ext_vector_type(2))) float v2f;
typedef __attribute__((ext_vector_type(8))) float v8f;
typedef __attribute__((ext_vector_type(4))) unsigned int v4ui;
typedef __attribute__((ext_vector_type(8))) int v8si;
typedef __attribute__((ext_vector_type(4))) int v4si;

#define HC 128          // channels for both layers (8x16 and 1x128)
#define NEG_SLOPE 0.2f

// Generic pointers to LDS carry the LDS byte offset in addr[31:0] (ISA flat
// addressing), so truncation yields the D# lds_addr.
__device__ __forceinline__ unsigned lds_offset_of(const void* p)
{
    return (unsigned)(unsigned long long)(uintptr_t)p;
}

// ---------------------------------------------------------------------------
// C[M x 128] = A[M x 128] * B[128 x 128] using V_WMMA_F32_16X16X4_F32.
// One wave per 16x16 output tile. The 128x16 B column-tile is staged into LDS
// with one Tensor Data Mover descriptor (tensor_load_to_lds + s_wait_tensorcnt),
// then the K loop feeds WMMA from LDS while A streams from global.
// A-frag (16x4 f32): lane m = lane&15; VGPR0 = K=(hi*2), VGPR1 = K=(hi*2)+1.
// B-frag (4x16 f32): lane n = lane&15; rows split the same way across halves.
// C/D (16x16 f32): VGPR r -> row (r + hi*8), col = lane&15.
// ---------------------------------------------------------------------------
__global__ __launch_bounds__(32)
void gat_gemm_wmma(const float* __restrict__ A, const float* __restrict__ B,
                   float* __restrict__ C, int M)
{
    __shared__ float Bs[HC * 16];       // B[:, tile_n .. tile_n+15], row-major

    const int lane   = threadIdx.x;
    const int tile_m = blockIdx.x * 16;
    const int tile_n = blockIdx.y * 16;
    const int l16    = lane & 15;
    const int hi     = lane >> 4;       // 0: lanes 0-15, 1: lanes 16-31
    const int kh     = hi * 2;          // K sub-offset inside each 4-wide step
    int m = tile_m + l16;
    if (m >= M) m = M - 1;              // clamp loads; stores are guarded below

    // ---- TDM: DMA the 128x16 f32 tile B[0:128, tile_n:tile_n+16] into LDS ----
    {
        const unsigned long long ga = (unsigned long long)(uintptr_t)(B + tile_n);
        v4ui g0;
        g0.x = 1u;                                    // count=1 valid user D#
        g0.y = lds_offset_of(Bs);                     // lds_addr (bytes)
        g0.z = (unsigned)ga;                          // global_addr[31:0]
        g0.w = (unsigned)((ga >> 32) & 0x01ffffffu)   // global_addr[56:32]
             | (2u << 30);                            // type = 2 ("image")
        v8si g1;
        g1[0] = 2 << 16;                              // data_size=2 -> 4-byte elems
        g1[1] = (int)((unsigned)HC << 16);            // tensor_dim0 = 128 ([63:48])
        g1[2] = (int)((unsigned)HC << 16);            // tensor_dim1 = 128 ([95:80])
        g1[3] = (int)(16u << 16);                     // tile_dim0   = 16  ([127:112])
        g1[4] = HC;                                   // tile_dim1 = 128, tile_dim2 = 0
        g1[5] = HC;                                   // tensor_dim0_stride = 128
        g1[6] = 0;                                    // stride hi / dim1_stride lo
        g1[7] = 0;
        const v4si z4 = {0, 0, 0, 0};                 // groups 2/3 unused (2D tile)
        const v8si z8 = {0, 0, 0, 0, 0, 0, 0, 0};
        __builtin_amdgcn_tensor_load_to_lds(g0, g1, z4, z4, z8, 0);
        __builtin_amdgcn_s_wait_tensorcnt(0);
    }

    const float* Arow = A + (size_t)m * HC;
    v8f acc = {};
#pragma unroll
    for (int k = 0; k < HC; k += 4) {
        v2f a, b;
        a.x = Arow[k + kh];
        a.y = Arow[k + kh + 1];
        b.x = Bs[(k + kh) * 16 + l16];
        b.y = Bs[(k + kh + 1) * 16 + l16];
        acc = __builtin_amdgcn_wmma_f32_16x16x4_f32(false, a, false, b,
                                                    (short)0, acc, false, false);
    }
    const int rbase = hi * 8;
#pragma unroll
    for (int r = 0; r < 8; ++r) {
        const int mo = tile_m + rbase + r;
        if (mo < M) C[(size_t)mo * HC + tile_n + l16] = acc[r];
    }
}

// ---------------------------------------------------------------------------
__global__ void gat_fill_f32(float* __restrict__ p, float v, long long n)
{
    long long i = (long long)blockIdx.x * blockDim.x + threadIdx.x;
    if (i < n) p[i] = v;
}

// alpha_src[n,h] = <h[n,h,:], a_src[h,:]> ; same for dst. One thread per (node,head).
__global__ void gat_alpha(const float* __restrict__ h,
                          const float* __restrict__ a_src, const float* __restrict__ a_dst,
                          float* __restrict__ as, float* __restrict__ ad,
                          int n, int heads, int outc)
{
    int i = blockIdx.x * blockDim.x + threadIdx.x;
    if (i >= n * heads) return;
    const int node = i / heads, hd = i - node * heads;
    const float* hp = h + (size_t)node * HC + hd * outc;
    const float* s  = a_src + (size_t)hd * outc;
    const float* d  = a_dst + (size_t)hd * outc;
    float accs = 0.f, accd = 0.f;
    for (int c = 0; c < outc; ++c) { float v = hp[c]; accs += v * s[c]; accd += v * d[c]; }
    as[i] = accs; ad[i] = accd;
}

__device__ __forceinline__ void edge_endpoints(const long long* __restrict__ ei,
                                               int E, int e, int& s, int& d)
{
    if (e < E) { s = (int)ei[e]; d = (int)ei[(size_t)E + e]; }
    else       { s = d = e - E; }                      // appended self-loops
}

// float atomic max via signed-max (val>=0) / unsigned-min (val<0) bit ordering
__device__ __forceinline__ void atomic_max_f32(float* addr, float val)
{
    if (val >= 0.f) atomicMax((int*)addr, __float_as_int(val));
    else            atomicMin((unsigned int*)addr, __float_as_uint(val));
}

__global__ void gat_edge_max(const long long* __restrict__ ei, int E, int n, int heads,
                             const float* __restrict__ as, const float* __restrict__ ad,
                             float* __restrict__ emax)
{
    int i = blockIdx.x * blockDim.x + threadIdx.x;
    if (i >= (E + n) * heads) return;
    const int e = i / heads, hd = i - e * heads;
    int s, d; edge_endpoints(ei, E, e, s, d);
    float v = as[s * heads + hd] + ad[d * heads + hd];
    v = v > 0.f ? v : NEG_SLOPE * v;                    // leaky_relu
    atomic_max_f32(&emax[d * heads + hd], v);
}

__global__ void gat_edge_sum(const long long* __restrict__ ei, int E, int n, int heads,
                             const float* __restrict__ as, const float* __restrict__ ad,
                             const float* __restrict__ emax, float* __restrict__ den)
{
    int i = blockIdx.x * blockDim.x + threadIdx.x;
    if (i >= (E + n) * heads) return;
    const int e = i / heads, hd = i - e * heads;
    int s, d; edge_endpoints(ei, E, e, s, d);
    float v = as[s * heads + hd] + ad[d * heads + hd];
    v = v > 0.f ? v : NEG_SLOPE * v;
    atomicAdd(&den[d * heads + hd], expf(v - emax[d * heads + hd]));
}

// 32 lanes per edge; each lane moves 4 contiguous channels (float4 gather,
// 4 float atomic-adds scatter). The edge id is wave-uniform, so readfirstlane
// lets the compiler keep edge endpoints and row bases in SGPRs (s_load path).
__global__ void gat_edge_aggr(const long long* __restrict__ ei, int E, int n,
                              int heads, int outc,
                              const float* __restrict__ h,
                              const float* __restrict__ as, const float* __restrict__ ad,
                              const float* __restrict__ emax, const float* __restrict__ den,
                              float* __restrict__ accum)
{
    long long gid = (long long)blockIdx.x * blockDim.x + threadIdx.x;
    int e = (int)(gid >> 5);
    e = __builtin_amdgcn_readfirstlane(e);             // wave-uniform edge id
    if (e >= E + n) return;
    const int lane = (int)(gid & 31);
    int s, d; edge_endpoints(ei, E, e, s, d);
    const int c0 = lane * 4;
    const int hd = c0 / outc;
    float v = as[s * heads + hd] + ad[d * heads + hd];
    v = v > 0.f ? v : NEG_SLOPE * v;
    const float alpha = expf(v - emax[d * heads + hd]) / den[d * heads + hd];
    const float4 hv = *(const float4*)(h + (size_t)s * HC + c0);
    float* o = accum + (size_t)d * HC + c0;
    atomicAdd(o + 0, hv.x * alpha);
    atomicAdd(o + 1, hv.y * alpha);
    atomicAdd(o + 2, hv.z * alpha);
    atomicAdd(o + 3, hv.w * alpha);
}

__global__ void gat_bias_elu(float* __restrict__ p, const float* __restrict__ b, long long n)
{
    long long i = (long long)blockIdx.x * blockDim.x + threadIdx.x;
    if (i >= n) return;
    float v = p[i] + b[i & (HC - 1)];
    p[i] = v > 0.f ? v : expm1f(v);                     // jax.nn.elu, alpha=1
}

__global__ void gat_bias_add(float* __restrict__ p, const float* __restrict__ b, long long n)
{
    long long i = (long long)blockIdx.x * blockDim.x + threadIdx.x;
    if (i >= n) return;
    p[i] += b[i & (HC - 1)];
}

// ---------------------------------------------------------------------------
extern "C" void kernel_launch(void* const* d_in, const int* in_sizes, int n_in,
                              void* d_out, int out_size, void* d_ws, size_t ws_size,
                              hipStream_t stream)
{
    const float*     x     = (const float*)d_in[0];
    const long long* ei    = (const long long*)d_in[1];   // int64 edge_index [2, E]
    const float*     W1    = (const float*)d_in[2];
    const float*     asrc1 = (const float*)d_in[3];
    const float*     adst1 = (const float*)d_in[4];
    const float*     b1    = (const float*)d_in[5];
    const float*     W2    = (const float*)d_in[6];
    const float*     asrc2 = (const float*)d_in[7];
    const float*     adst2 = (const float*)d_in[8];
    const float*     b2    = (const float*)d_in[9];
    float* out = (float*)d_out;

    const int N  = in_sizes[0] / HC;
    const int E  = in_sizes[1] / 2;
    const int EN = E + N;

    float* ws = (float*)d_ws;
    const size_t nh = (size_t)N * HC;
    float* h1   = ws;                    // x @ W1                [N,128]
    float* x2   = ws + nh;               // layer-1 accum -> ELU  [N,128]
    float* h2   = ws + 2 * nh;           // x2 @ W2               [N,128]
    float* asb  = ws + 3 * nh;           // alpha_src             [N,8]
    float* adb  = asb + (size_t)N * 8;   // alpha_dst             [N,8]
    float* emax = adb + (size_t)N * 8;   // segment max           [N,8]
    float* den  = emax + (size_t)N * 8;  // segment sum           [N,8]

    const int TB = 256;
    auto nb = [](long long n, int tb) { return (unsigned)((n + tb - 1) / tb); };
    const dim3 ggrid((N + 15) / 16, HC / 16);

    // ---------------- layer 1: 8 heads x 16 channels, concat ----------------
    gat_fill_f32<<<nb((long long)N * 8, TB), TB, 0, stream>>>(emax, -FLT_MAX, (long long)N * 8);
    gat_fill_f32<<<nb((long long)N * 8, TB), TB, 0, stream>>>(den, 0.f, (long long)N * 8);
    gat_fill_f32<<<nb((long long)nh, TB), TB, 0, stream>>>(x2, 0.f, (long long)nh);
    gat_fill_f32<<<nb((long long)nh, TB), TB, 0, stream>>>(out, 0.f, (long long)nh);

    gat_gemm_wmma<<<ggrid, 32, 0, stream>>>(x, W1, h1, N);
    gat_alpha<<<nb((long long)N * 8, TB), TB, 0, stream>>>(h1, asrc1, adst1, asb, adb, N, 8, 16);
    gat_edge_max<<<nb((long long)EN * 8, TB), TB, 0, stream>>>(ei, E, N, 8, asb, adb, emax);
    gat_edge_sum<<<nb((long long)EN * 8, TB), TB, 0, stream>>>(ei, E, N, 8, asb, adb, emax, den);
    gat_edge_aggr<<<nb((long long)EN * 32, TB), TB, 0, stream>>>(ei, E, N, 8, 16,
                                                                 h1, asb, adb, emax, den, x2);
    gat_bias_elu<<<nb((long long)nh, TB), TB, 0, stream>>>(x2, b1, (long long)nh);

    // ---------------- layer 2: 1 head x 128 channels, mean == identity ------
    gat_fill_f32<<<nb((long long)N, TB), TB, 0, stream>>>(emax, -FLT_MAX, (long long)N);
    gat_fill_f32<<<nb((long long)N, TB), TB, 0, stream>>>(den, 0.f, (long long)N);

    gat_gemm_wmma<<<ggrid, 32, 0, stream>>>(x2, W2, h2, N);
    gat_alpha<<<nb((long long)N, TB), TB, 0, stream>>>(h2, asrc2, adst2, asb, adb, N, 1, HC);
    gat_edge_max<<<nb((long long)EN, TB), TB, 0, stream>>>(ei, E, N, 1, asb, adb, emax);
    gat_edge_sum<<<nb((long long)EN, TB), TB, 0, stream>>>(ei, E, N, 1, asb, adb, emax, den);
    gat_edge_aggr<<<nb((long long)EN * 32, TB), TB, 0, stream>>>(ei, E, N, 1, HC,
                                                                 h2, asb, adb, emax, den, out);
    gat_bias_add<<<nb((long long)nh, TB), TB, 0, stream>>>(out, b2, (long long)nh);
}